// ContextFusion_53334903882146
// MI455X (gfx1250) — compile-verified
//
#include <hip/hip_runtime.h>

#define EPS_F 1e-5f
#define BB 2
#define NN 1024
#define CC 64          // C
#define KK (2 * CC)    // 2C = 128
#define FC_STRIDE 132  // padded LDS row stride (floats) to avoid bank conflicts

typedef __attribute__((ext_vector_type(2))) float v2f;
typedef __attribute__((ext_vector_type(8))) float v8f;

// ---------------------------------------------------------------------------
// Kernel 1: per-(b,c) statistics.
//   s[b,c]   = gamma[c] * rsqrt(var_j x[b,j,c] + eps)
//   m[b,c]   = sum_j softmax_j(s*x[b,j,c]) * x[b,j,c]
// One block per batch; thread t handles channel c = t&63, row-group g = t>>6.
// ---------------------------------------------------------------------------
__global__ __launch_bounds__(256) void cf_stats_kernel(
    const float* __restrict__ x, const float* __restrict__ gamma,
    float2* __restrict__ stats) {
  const int b = blockIdx.x;
  const int tid = (int)threadIdx.x;
  const int c = tid & 63;
  const int g = tid >> 6;  // 0..3
  const float* xb = x + (size_t)b * NN * CC;

  float sum = 0.f, sumsq = 0.f;
  float vmax = -3.4e38f, vmin = 3.4e38f;
  for (int j = g; j < NN; j += 4) {
    float v = xb[j * CC + c];  // consecutive c across lanes -> coalesced
    sum += v;
    sumsq += v * v;
    vmax = fmaxf(vmax, v);
    vmin = fminf(vmin, v);
  }

  __shared__ float r0[256], r1[256], r2[256], r3[256];
  __shared__ float sS[64], sL[64];
  r0[tid] = sum; r1[tid] = sumsq; r2[tid] = vmax; r3[tid] = vmin;
  __syncthreads();
  if (tid < 64) {
    float s  = r0[tid] + r0[tid + 64] + r0[tid + 128] + r0[tid + 192];
    float sq = r1[tid] + r1[tid + 64] + r1[tid + 128] + r1[tid + 192];
    float mx = fmaxf(fmaxf(r2[tid], r2[tid + 64]),
                     fmaxf(r2[tid + 128], r2[tid + 192]));
    float mn = fminf(fminf(r3[tid], r3[tid + 64]),
                     fminf(r3[tid + 128], r3[tid + 192]));
    float mean = s * (1.f / NN);
    float var = sq * (1.f / NN) - mean * mean;  // biased (ddof=0), matches jnp.var
    float scl = gamma[tid] * rsqrtf(var + EPS_F);
    sS[tid] = scl;
    // max over j of logit scl*x: handle either sign of scl
    sL[tid] = fmaxf(scl * mx, scl * mn);
  }
  __syncthreads();

  const float scl = sS[c];
  const float lmax = sL[c];
  float Z = 0.f, M = 0.f;
  for (int j = g; j < NN; j += 4) {
    float v = xb[j * CC + c];
    float e = __expf(scl * v - lmax);
    Z += e;
    M += e * v;
  }
  r0[tid] = Z; r1[tid] = M;
  __syncthreads();
  if (tid < 64) {
    float Zt = r0[tid] + r0[tid + 64] + r0[tid + 128] + r0[tid + 192];
    float Mt = r1[tid] + r1[tid + 64] + r1[tid + 128] + r1[tid + 192];
    stats[b * CC + tid] = make_float2(sS[tid], Mt / Zt);
  }
}

// ---------------------------------------------------------------------------
// Kernel 2: build fc tile (16 rows x 128) in LDS, GEMM vs W^T with f32 WMMA,
// fused ReLU. One block = 4 waves; wave w owns output cols [16w, 16w+16).
//   out[r, n] = relu( sum_k fc[r,k] * W[n,k] )
// ---------------------------------------------------------------------------
__global__ __launch_bounds__(128) void cf_gemm_kernel(
    const float* __restrict__ x, const float* __restrict__ gamma,
    const float* __restrict__ beta, const float* __restrict__ W,
    const float2* __restrict__ stats, float* __restrict__ out) {
  __shared__ float fcS[16 * FC_STRIDE];

  const int strip = blockIdx.x;    // 0..127
  const int row0 = strip * 16;     // global row (b*N + i), strip fits in one b
  const int b = strip >> 6;        // 16*64 = 1024 rows per batch
  const int tid = (int)threadIdx.x;

  // Stage fc: fc[r, c] = s*(m - x) + beta[c] ; fc[r, C+c] = gamma[C+c]*x + beta[C+c]
  for (int idx = tid; idx < 16 * CC; idx += 128) {
    int r = idx >> 6;
    int c = idx & 63;
    float v = x[(size_t)(row0 + r) * CC + c];
    float2 sm = stats[b * CC + c];
    fcS[r * FC_STRIDE + c] = sm.x * (sm.y - v) + beta[c];
    fcS[r * FC_STRIDE + CC + c] = gamma[CC + c] * v + beta[CC + c];
  }
  __syncthreads();

  const int wave = tid >> 5;            // 0..3 -> column tile
  const int lane = tid & 31;
  const int half = lane >> 4;           // 0: lanes 0-15, 1: lanes 16-31
  const int mrow = lane & 15;           // A-matrix row (M) for this lane
  const int ncol = lane & 15;           // B/D column (N) within tile
  const int khalf = half << 1;          // K sub-offset {0,2} (ISA 16x4 f32 layout)
  const float* wrow = W + (size_t)(wave * 16 + ncol) * KK;  // W[n, :] = Wt[:, n]

  v8f acc = {};
#pragma unroll 4
  for (int kb = 0; kb < KK; kb += 4) {
    // A frag: lane holds A[mrow][kb+khalf], A[mrow][kb+khalf+1]
    v2f a = *(const v2f*)&fcS[mrow * FC_STRIDE + kb + khalf];
    // B frag: lane holds Wt[kb+khalf][n], Wt[kb+khalf+1][n] = W[n][kb+khalf..+1]
    v2f bf = *(const v2f*)&wrow[kb + khalf];
    acc = __builtin_amdgcn_wmma_f32_16x16x4_f32(
        /*neg_a=*/false, a, /*neg_b=*/false, bf,
        /*c_mod=*/(short)0, acc, /*reuse_a=*/false, /*reuse_b=*/false);
  }

  // D layout: acc[r] -> (M = r + 8*half, N = ncol); fused ReLU store.
  const int mbase = half << 3;
  float* obase = out + (size_t)row0 * CC + wave * 16 + ncol;
#pragma unroll
  for (int r = 0; r < 8; ++r) {
    obase[(size_t)(mbase + r) * CC] = fmaxf(acc[r], 0.f);
  }
}

extern "C" void kernel_launch(void* const* d_in, const int* in_sizes, int n_in,
                              void* d_out, int out_size, void* d_ws, size_t ws_size,
                              hipStream_t stream) {
  const float* x = (const float*)d_in[0];      // [B, N, C]
  const float* gamma = (const float*)d_in[1];  // [2C]
  const float* beta = (const float*)d_in[2];   // [2C]
  const float* W = (const float*)d_in[3];      // [C, 2C]
  float* out = (float*)d_out;                  // [B, N, C]
  float2* stats = (float2*)d_ws;               // [B*C] of (s, m)

  cf_stats_kernel<<<BB, 256, 0, stream>>>(x, gamma, stats);
  cf_gemm_kernel<<<(BB * NN) / 16, 128, 0, stream>>>(x, gamma, beta, W, stats, out);
}